// GATRecommender_73572789780483
// MI455X (gfx1250) — compile-verified
//
#include <hip/hip_runtime.h>
#include <math.h>

#define NU 200000
#define NS 500000
#define NA 50000
#define NN 700000      // NU + NS
#define DD 128
#define HH 4
#define CC 32
#define EE 2000000
#define NEG_SLOPE 0.2f
#define EPSV 1e-16f

#define RT1 7          // row tiles / block in GEMM1: 112 rows, 700000/112 = 6250
#define RT2 5          // row tiles / block in GEMM2: 80 rows, 750000/80 = 9375

typedef __attribute__((ext_vector_type(2))) float v2f;
typedef __attribute__((ext_vector_type(8))) float v8f;

// float atomic max via int/uint punning (correct for mixed-sign IEEE floats)
__device__ __forceinline__ void atomicMaxF(float* addr, float v) {
    if (v >= 0.0f) atomicMax((int*)addr, __float_as_int(v));
    else           atomicMin((unsigned int*)addr, __float_as_uint(v));
}

// ---------------- init: agg <- gat_bias, segmax <- -inf, den <- EPS ----------
__global__ void k_init(float* __restrict__ outagg, const float* __restrict__ gat_bias,
                       float* __restrict__ segmax, float* __restrict__ den) {
    size_t i = (size_t)blockIdx.x * blockDim.x + threadIdx.x;
    if (i < (size_t)NN * DD) outagg[i] = gat_bias[i & (DD - 1)];
    if (i < (size_t)NN * HH) { segmax[i] = -__builtin_inff(); den[i] = EPSV; }
}

// -------- fold attention vectors: ws[k,h] = sum_c gat_w[k, h*32+c]*att_src[h,c]
__global__ void k_attw(const float* __restrict__ gat_w,
                       const float* __restrict__ att_src, const float* __restrict__ att_dst,
                       float* __restrict__ wsv, float* __restrict__ wdv) {
    int t = threadIdx.x;            // 512 threads
    int k = t >> 2, h = t & 3;
    float s = 0.f, d = 0.f;
    for (int c = 0; c < CC; ++c) {
        float w = gat_w[k * DD + h * CC + c];
        s += w * att_src[h * CC + c];
        d += w * att_dst[h * CC + c];
    }
    wsv[k * HH + h] = s;
    wdv[k * HH + h] = d;
}

// ---------------- GEMM1: xt = x_cat @ gat_w ; fused a_s, a_d ----------------
// 256 threads = 8 waves; each wave owns one 16-col tile with the B panel held
// in 64 VGPRs (loaded once per block); 7 row tiles of 16 per block.
__global__ __launch_bounds__(256) void k_gat_gemm(
    const int* __restrict__ user_ids, const int* __restrict__ song_ids,
    const float* __restrict__ user_emb, const float* __restrict__ song_emb,
    const float* __restrict__ gat_w,
    const float* __restrict__ wsv, const float* __restrict__ wdv,
    float* __restrict__ xt, float* __restrict__ a_s, float* __restrict__ a_d) {
    __shared__ float As[RT1 * 16][132];   // 112x132 floats = 59 KB, rows 16B aligned
    const int t  = threadIdx.x;
    const int m0 = blockIdx.x * (RT1 * 16);

    // stage A: per row tile, 16 rows x 128 cols, 8 floats per thread
    {
        int r  = t >> 4;            // 0..15
        int c0 = (t & 15) * 8;      // 0..120
        #pragma unroll
        for (int i = 0; i < RT1; ++i) {
            int g = m0 + i * 16 + r;
            const float* src = (g < NU) ? (user_emb + (size_t)user_ids[g] * DD)
                                        : (song_emb + (size_t)song_ids[g - NU] * DD);
            const float4* s4 = (const float4*)(src + c0);
            float4* d4 = (float4*)&As[i * 16 + r][c0];
            d4[0] = s4[0];
            d4[1] = s4[1];
        }
    }

    const int lane = t & 31;
    const int wv   = t >> 5;                 // wave 0..7, each owns 16 output cols
    const int row  = lane & 15;
    const int kOff = (lane >> 4) << 1;       // upper half-lanes take K+2,K+3
    const int col  = wv * 16 + (lane & 15);
    const int mOff = (lane >> 4) << 3;       // C/D: VGPR j -> M = j + 8*(lane>=16)

    // hoist this wave's B panel into registers: 64 floats per lane
    float breg[64];
    #pragma unroll
    for (int kb = 0; kb < DD; kb += 4) {
        breg[(kb >> 1) + 0] = gat_w[(size_t)(kb + kOff) * DD + col];
        breg[(kb >> 1) + 1] = gat_w[(size_t)(kb + kOff + 1) * DD + col];
    }
    __syncthreads();

    #pragma unroll
    for (int i = 0; i < RT1; ++i) {
        const int r0 = i * 16;
        v8f acc = {};
        #pragma unroll
        for (int kb = 0; kb < DD; kb += 4) {
            v2f a, b;
            a[0] = As[r0 + row][kb + kOff];
            a[1] = As[r0 + row][kb + kOff + 1];
            b[0] = breg[(kb >> 1) + 0];
            b[1] = breg[(kb >> 1) + 1];
            acc = __builtin_amdgcn_wmma_f32_16x16x4_f32(
                false, a, false, b, (short)0, acc, false, false);
        }
        const int mrow = m0 + r0 + mOff;
        #pragma unroll
        for (int j = 0; j < 8; ++j)
            xt[(size_t)(mrow + j) * DD + col] = acc[j];
    }

    // fused attention dots: 112 rows x {a_s,a_d} x 4 heads
    {
        int q = t & 7;
        int h = q & 3;
        const float* wcol = (q < 4) ? wsv : wdv;
        float* dst = (q < 4) ? a_s : a_d;
        for (int r = t >> 3; r < RT1 * 16; r += 32) {
            float sum = 0.f;
            #pragma unroll 8
            for (int k = 0; k < DD; ++k) sum += As[r][k] * wcol[k * HH + h];
            dst[(size_t)(m0 + r) * HH + h] = sum;
        }
    }
}

// ---------------- edge pass 1: per-destination max of leaky-relu logits ------
__global__ void k_edge_max(const int* __restrict__ esrc, const int* __restrict__ edst,
                           const float* __restrict__ a_s, const float* __restrict__ a_d,
                           float* __restrict__ segmax) {
    int e = blockIdx.x * blockDim.x + threadIdx.x;
    if (e >= EE) return;
    int s = esrc[e], d = edst[e];
    #pragma unroll
    for (int h = 0; h < HH; ++h) {
        float x = a_s[(size_t)s * HH + h] + a_d[(size_t)d * HH + h];
        float logit = x > 0.f ? x : NEG_SLOPE * x;
        atomicMaxF(&segmax[(size_t)d * HH + h], logit);
    }
}

// ---------------- edge pass 2: softmax denominators --------------------------
__global__ void k_edge_den(const int* __restrict__ esrc, const int* __restrict__ edst,
                           const float* __restrict__ a_s, const float* __restrict__ a_d,
                           const float* __restrict__ segmax, float* __restrict__ den) {
    int e = blockIdx.x * blockDim.x + threadIdx.x;
    if (e >= EE) return;
    int s = esrc[e], d = edst[e];
    #pragma unroll
    for (int h = 0; h < HH; ++h) {
        float x = a_s[(size_t)s * HH + h] + a_d[(size_t)d * HH + h];
        float logit = x > 0.f ? x : NEG_SLOPE * x;
        float ex = expf(logit - segmax[(size_t)d * HH + h]);
        atomicAdd(&den[(size_t)d * HH + h], ex);
    }
}

// ---------------- edge pass 3: weighted message scatter (1 wave / edge) ------
__global__ __launch_bounds__(256) void k_edge_msg(
    const int* __restrict__ esrc, const int* __restrict__ edst,
    const float* __restrict__ a_s, const float* __restrict__ a_d,
    const float* __restrict__ segmax, const float* __restrict__ den,
    const float* __restrict__ xt, float* __restrict__ outagg) {
    int e    = blockIdx.x * 8 + (threadIdx.x >> 5);
    int lane = threadIdx.x & 31;
    if (e >= EE) return;
    int s = esrc[e], d = edst[e];
    float av = 0.f;
    if (lane < HH) {
        float x = a_s[(size_t)s * HH + lane] + a_d[(size_t)d * HH + lane];
        float logit = x > 0.f ? x : NEG_SLOPE * x;
        av = expf(logit - segmax[(size_t)d * HH + lane]) / den[(size_t)d * HH + lane];
    }
    #pragma unroll
    for (int h = 0; h < HH; ++h) {
        float ah  = __shfl(av, h, 32);
        float val = xt[(size_t)s * DD + h * CC + lane] * ah;
        atomicAdd(&outagg[(size_t)d * DD + h * CC + lane], val);
    }
}

// ---------------- GEMM2: out = (relu(agg)+x_old) @ proj_w + proj_b -----------
__global__ __launch_bounds__(256) void k_proj(
    const int* __restrict__ user_ids, const int* __restrict__ song_ids,
    const int* __restrict__ artist_ids,
    const float* __restrict__ user_emb, const float* __restrict__ song_emb,
    const float* __restrict__ artist_emb,
    const float* __restrict__ proj_w, const float* __restrict__ proj_b,
    float* __restrict__ out) {
    __shared__ float As[RT2 * 16][132];   // 80x132 floats = 42 KB
    const int t  = threadIdx.x;
    const int m0 = blockIdx.x * (RT2 * 16);

    // stage relu(xnew)+xold (or relu(xa)+xa for artists)
    {
        int r  = t >> 4;
        int c0 = (t & 15) * 8;
        #pragma unroll
        for (int i = 0; i < RT2; ++i) {
            int g = m0 + i * 16 + r;
            float tmp[8];
            if (g < NN) {
                const float* xo = (g < NU) ? (user_emb + (size_t)user_ids[g] * DD)
                                           : (song_emb + (size_t)song_ids[g - NU] * DD);
                const float* xn = out + (size_t)g * DD;
                #pragma unroll
                for (int j = 0; j < 8; ++j) {
                    float v = xn[c0 + j];
                    tmp[j] = (v > 0.f ? v : 0.f) + xo[c0 + j];
                }
            } else {
                const float* xa = artist_emb + (size_t)artist_ids[g - NN] * DD;
                #pragma unroll
                for (int j = 0; j < 8; ++j) {
                    float v = xa[c0 + j];
                    tmp[j] = (v > 0.f ? v : 0.f) + v;   // relu(xa) + xa
                }
            }
            #pragma unroll
            for (int j = 0; j < 8; ++j) As[i * 16 + r][c0 + j] = tmp[j];
        }
    }

    const int lane = t & 31;
    const int wv   = t >> 5;
    const int row  = lane & 15;
    const int kOff = (lane >> 4) << 1;
    const int col  = wv * 16 + (lane & 15);
    const int mOff = (lane >> 4) << 3;

    float breg[64];
    #pragma unroll
    for (int kb = 0; kb < DD; kb += 4) {
        breg[(kb >> 1) + 0] = proj_w[(size_t)(kb + kOff) * DD + col];
        breg[(kb >> 1) + 1] = proj_w[(size_t)(kb + kOff + 1) * DD + col];
    }
    const float bias = proj_b[col];
    __syncthreads();

    #pragma unroll
    for (int i = 0; i < RT2; ++i) {
        const int r0 = i * 16;
        v8f acc = {};
        #pragma unroll
        for (int kb = 0; kb < DD; kb += 4) {
            v2f a, b;
            a[0] = As[r0 + row][kb + kOff];
            a[1] = As[r0 + row][kb + kOff + 1];
            b[0] = breg[(kb >> 1) + 0];
            b[1] = breg[(kb >> 1) + 1];
            acc = __builtin_amdgcn_wmma_f32_16x16x4_f32(
                false, a, false, b, (short)0, acc, false, false);
        }
        const int mrow = m0 + r0 + mOff;
        #pragma unroll
        for (int j = 0; j < 8; ++j)
            out[(size_t)(mrow + j) * DD + col] = acc[j] + bias;
    }
}

extern "C" void kernel_launch(void* const* d_in, const int* in_sizes, int n_in,
                              void* d_out, int out_size, void* d_ws, size_t ws_size,
                              hipStream_t stream) {
    const int*   user_ids   = (const int*)d_in[0];
    const int*   song_ids   = (const int*)d_in[1];
    const int*   artist_ids = (const int*)d_in[2];
    const int*   edge_src   = (const int*)d_in[3];
    const int*   edge_dst   = (const int*)d_in[4];
    const float* user_emb   = (const float*)d_in[5];
    const float* song_emb   = (const float*)d_in[6];
    const float* artist_emb = (const float*)d_in[7];
    const float* gat_w      = (const float*)d_in[8];
    const float* att_src    = (const float*)d_in[9];
    const float* att_dst    = (const float*)d_in[10];
    const float* gat_bias   = (const float*)d_in[11];
    const float* proj_w     = (const float*)d_in[12];
    const float* proj_b     = (const float*)d_in[13];
    float* out = (float*)d_out;

    // workspace carve-up (floats)
    float* W = (float*)d_ws;
    size_t off = 0;
    float* xt     = W + off; off += (size_t)NN * DD;   // 89.6M floats
    float* a_s    = W + off; off += (size_t)NN * HH;
    float* a_d    = W + off; off += (size_t)NN * HH;
    float* segmax = W + off; off += (size_t)NN * HH;
    float* den    = W + off; off += (size_t)NN * HH;
    float* wsv    = W + off; off += (size_t)DD * HH;
    float* wdv    = W + off; off += (size_t)DD * HH;

    k_init<<<((size_t)NN * DD + 255) / 256, 256, 0, stream>>>(out, gat_bias, segmax, den);
    k_attw<<<1, 512, 0, stream>>>(gat_w, att_src, att_dst, wsv, wdv);
    k_gat_gemm<<<NN / (RT1 * 16), 256, 0, stream>>>(user_ids, song_ids, user_emb, song_emb,
                                                    gat_w, wsv, wdv, xt, a_s, a_d);
    k_edge_max<<<(EE + 255) / 256, 256, 0, stream>>>(edge_src, edge_dst, a_s, a_d, segmax);
    k_edge_den<<<(EE + 255) / 256, 256, 0, stream>>>(edge_src, edge_dst, a_s, a_d, segmax, den);
    k_edge_msg<<<EE / 8, 256, 0, stream>>>(edge_src, edge_dst, a_s, a_d, segmax, den, xt, out);
    k_proj<<<(NN + NA) / (RT2 * 16), 256, 0, stream>>>(user_ids, song_ids, artist_ids,
                                                       user_emb, song_emb, artist_emb,
                                                       proj_w, proj_b, out);
}